// AngularPenaltySMLoss_51582557225658
// MI455X (gfx1250) — compile-verified
//
#include <hip/hip_runtime.h>
#include <hip/hip_bf16.h>
#include <stdint.h>

// CosFace loss, MI455X (gfx1250). Memory-bound: 2048x50257 f32 = 412 MB read,
// floor ~17.7us @ 23.3 TB/s. Per-wave async global->LDS streaming pipeline
// (8 stages, NT cache hint: data is touched exactly once), WMMA wave reduce.

#define S_SCALE   30.0f
#define MARGIN    0.4f
#define N_ROWS    2048
#define N_CLASSES 50257

#define KS    43.28085122666891f     // S * log2(e)
#define LOG2E 1.4426950408889634f
#define LN2   0.6931471805599453f

#define WAVES  8
#define TPB    256
#define STAGES 8                     // 8 x 512B in flight per wave
#define NFULL  (N_CLASSES / 128)     // 392 full 128-float iterations per row

typedef __attribute__((ext_vector_type(2))) float v2f;
typedef __attribute__((ext_vector_type(8))) float v8f;

__device__ __forceinline__ void async_ld_b128_nt(uint32_t lds_byte_off,
                                                 uint32_t glb_byte_off,
                                                 uint64_t base) {
    // GVS mode: the 64-bit SGPR base plus the 32-bit VGPR offset gives the global address; copies 16 bytes per lane from it to the LDS address in vdst
    // NT: single-use streaming data, mark for early eviction (192MB L2, 412MB stream).
    asm volatile("global_load_async_to_lds_b128 %0, %1, %2 th:TH_LOAD_NT"
                 :
                 : "v"(lds_byte_off), "v"(glb_byte_off), "s"(base)
                 : "memory");
}

__global__ __launch_bounds__(TPB) void cosface_rowloss_kernel(
        const float* __restrict__ x,
        const int* __restrict__ labels,
        float* __restrict__ rowloss) {
    __shared__ __align__(16) float tile[WAVES][STAGES][128];
    __shared__ float wsum_lds[WAVES];

    const int row  = blockIdx.x;
    const int tid  = threadIdx.x;
    const int wave = tid >> 5;
    const int lane = tid & 31;

    const float* rowp = x + (size_t)row * N_CLASSES;
    const uint64_t base = (uint64_t)(uintptr_t)rowp;

    // Per-lane LDS byte offsets for this wave's pipeline slots.
    uint32_t ldsOff[STAGES];
#pragma unroll
    for (int s = 0; s < STAGES; ++s)
        ldsOff[s] = (uint32_t)(uintptr_t)(&tile[wave][s][lane * 4]);

    // ---- prologue: fill pipeline (NFULL=392 >> STAGES*WAVES=64, always safe) ----
#pragma unroll
    for (int s = 0; s < STAGES; ++s) {
        const int it = wave + s * WAVES;
        async_ld_b128_nt(ldsOff[s], (uint32_t)(it * 512 + lane * 16), base);
    }

    float acc = 0.0f;
    int i = wave;
    int slot = 0;

    // ---- steady state: wait oldest, consume, reissue into freed slot ----
    for (int j = wave + STAGES * WAVES; j < NFULL; j += WAVES, i += WAVES) {
        asm volatile("s_wait_asynccnt 7" ::: "memory");   // STAGES-1: oldest done
        const float4 v = *(const float4*)&tile[wave][slot][lane * 4];
        asm volatile("s_wait_dscnt 0" ::: "memory");      // ds_load drained before slot reuse
        async_ld_b128_nt(ldsOff[slot], (uint32_t)(j * 512 + lane * 16), base);
        slot = (slot + 1) & (STAGES - 1);
        acc += __builtin_amdgcn_exp2f(v.x * KS);
        acc += __builtin_amdgcn_exp2f(v.y * KS);
        acc += __builtin_amdgcn_exp2f(v.z * KS);
        acc += __builtin_amdgcn_exp2f(v.w * KS);
    }

    // ---- drain remaining STAGES tiles ----
    asm volatile("s_wait_asynccnt 0" ::: "memory");
    for (; i < NFULL; i += WAVES) {
        const float4 v = *(const float4*)&tile[wave][slot][lane * 4];
        slot = (slot + 1) & (STAGES - 1);
        acc += __builtin_amdgcn_exp2f(v.x * KS);
        acc += __builtin_amdgcn_exp2f(v.y * KS);
        acc += __builtin_amdgcn_exp2f(v.z * KS);
        acc += __builtin_amdgcn_exp2f(v.w * KS);
    }

    // ---- tail columns (50257 - 392*128 = 81) via plain loads ----
    {
        const int c = NFULL * 128 + tid;
        if (c < N_CLASSES)
            acc += __builtin_amdgcn_exp2f(rowp[c] * KS);
    }

    // ---- wave reduction via V_WMMA_F32_16X16X4_F32 (B = ones) ----
    // A(16x4): lanes0-15 {K0,K1}, lanes16-31 {K2,K3}. acc in .x, 0 in .y:
    // D[m,*] = acc[lane m] + acc[lane m+16]. Lanes<16 hold D rows 0..7,
    // lanes>=16 hold rows 8..15 -> sum 8 regs, then shfl_xor(16).
    v2f a; a[0] = acc;  a[1] = 0.0f;
    v2f b; b[0] = 1.0f; b[1] = 1.0f;
    v8f c = {};
    c = __builtin_amdgcn_wmma_f32_16x16x4_f32(false, a, false, b,
                                              (short)0, c, false, false);
    float h = ((c[0] + c[1]) + (c[2] + c[3])) + ((c[4] + c[5]) + (c[6] + c[7]));
    float wsum = h + __shfl_xor(h, 16, 32);

    if (lane == 0) wsum_lds[wave] = wsum;
    __syncthreads();

    if (tid == 0) {
        float sumExp = 0.0f;
#pragma unroll
        for (int w = 0; w < WAVES; ++w) sumExp += wsum_lds[w];

        const int lbl = labels[row];
        const float ll = rowp[lbl];
        const float numer = S_SCALE * (ll - MARGIN);
        const float excl  = sumExp - __builtin_amdgcn_exp2f(ll * KS);
        const float denom = __builtin_amdgcn_exp2f(numer * LOG2E) + excl;
        rowloss[row] = numer - __builtin_amdgcn_logf(denom) * LN2;
    }
}

__global__ __launch_bounds__(TPB) void cosface_final_reduce(
        const float* __restrict__ rowloss,
        float* __restrict__ out) {
    __shared__ float ws[WAVES];
    float a = 0.0f;
    for (int i = threadIdx.x; i < N_ROWS; i += TPB) a += rowloss[i];
#pragma unroll
    for (int o = 16; o > 0; o >>= 1) a += __shfl_down(a, o, 32);
    if ((threadIdx.x & 31) == 0) ws[threadIdx.x >> 5] = a;
    __syncthreads();
    if (threadIdx.x == 0) {
        float t = 0.0f;
#pragma unroll
        for (int w = 0; w < WAVES; ++w) t += ws[w];
        *out = -t * (1.0f / (float)N_ROWS);
    }
}

extern "C" void kernel_launch(void* const* d_in, const int* in_sizes, int n_in,
                              void* d_out, int out_size, void* d_ws, size_t ws_size,
                              hipStream_t stream) {
    const float* x      = (const float*)d_in[0];
    const int*   labels = (const int*)d_in[1];   // harness contract: integer -> const int*
    float* rowloss = (float*)d_ws;               // 2048 floats of scratch
    (void)in_sizes; (void)n_in; (void)out_size; (void)ws_size;

    cosface_rowloss_kernel<<<N_ROWS, TPB, 0, stream>>>(x, labels, rowloss);
    cosface_final_reduce<<<1, TPB, 0, stream>>>(rowloss, (float*)d_out);
}